// AnchorTargetLayer_82678120448579
// MI455X (gfx1250) — compile-verified
//
#include <hip/hip_runtime.h>
#include <stdint.h>

// ---------------- problem constants (match reference) ----------------
#define H_ 192
#define W_ 192
#define A_ 9
#define N_ANCH (H_ * W_ * A_)      // 331776 == 1296 * 256
#define NUM_GT 128
#define RPN_NEG 0.3f
#define RPN_POS 0.7f
#define RPN_BATCH 256
#define NUM_FG 128
#define FG_CAP 32768
#define TIE_CAP 32768
#define NBINS 1024
#define NBLK (N_ANCH / 256)

// counters layout in workspace
#define CNT_FG 0
#define CNT_TIE 1
#define CNT_FGKEPT 2
#define CNT_NUMBG 3
#define CNT_NUMEX 4
#define CNT_BOUND 5
#define CNT_RANKBASE 6

// workspace byte offsets
#define OFF_GTMAX 0                         // 128 * u32
#define OFF_CNT   512                       // 8 * i32
#define OFF_HIST  1024                      // NBINS * i32
#define OFF_MAXOV 8192                      // N * f32
#define OFF_ARG   (8192 + 4 * N_ANCH)       // N * i32
#define OFF_FG    (8192 + 8 * N_ANCH)       // FG_CAP * uint2
#define OFF_TIE   (OFF_FG + 8 * FG_CAP)     // TIE_CAP * uint2

typedef __attribute__((ext_vector_type(4))) unsigned int u32x4;
typedef __attribute__((ext_vector_type(8))) int i32x8;
typedef __attribute__((ext_vector_type(4))) int i32x4;

// ---- monotone float<->uint encoding so atomicMax(unsigned) orders floats ----
__device__ __forceinline__ unsigned fenc(float f) {
  unsigned b = __float_as_uint(f);
  return (b & 0x80000000u) ? ~b : (b | 0x80000000u);
}
__device__ __forceinline__ float fdec(unsigned u) {
  return (u & 0x80000000u) ? __uint_as_float(u & 0x7fffffffu) : __uint_as_float(~u);
}

// ---------------- Threefry2x32-20 (same ARX structure as jax.random) --------
__device__ __forceinline__ void tfr(uint32_t& x0, uint32_t& x1, int r) {
  x0 += x1; x1 = (x1 << r) | (x1 >> (32 - r)); x1 ^= x0;
}
__device__ uint2 threefry2x32(uint32_t k0, uint32_t k1, uint32_t c0, uint32_t c1) {
  uint32_t ks2 = k0 ^ k1 ^ 0x1BD11BDAu;
  uint32_t x0 = c0 + k0, x1 = c1 + k1;
  tfr(x0,x1,13); tfr(x0,x1,15); tfr(x0,x1,26); tfr(x0,x1,6);  x0 += k1;  x1 += ks2 + 1u;
  tfr(x0,x1,17); tfr(x0,x1,29); tfr(x0,x1,16); tfr(x0,x1,24); x0 += ks2; x1 += k0 + 2u;
  tfr(x0,x1,13); tfr(x0,x1,15); tfr(x0,x1,26); tfr(x0,x1,6);  x0 += k0;  x1 += k1 + 3u;
  tfr(x0,x1,17); tfr(x0,x1,29); tfr(x0,x1,16); tfr(x0,x1,24); x0 += k1;  x1 += ks2 + 4u;
  tfr(x0,x1,13); tfr(x0,x1,15); tfr(x0,x1,26); tfr(x0,x1,6);  x0 += ks2; x1 += k0 + 5u;
  return make_uint2(x0, x1);
}
__device__ __forceinline__ float u01(uint32_t bits) {
  return __uint_as_float(0x3f800000u | (bits >> 9)) - 1.0f;
}
__device__ __forceinline__ float draw_uniform(uint32_t c0, uint32_t c1, uint32_t idx) {
  // key derived from jax.random.key(42) split; per-index counter
  uint2 k = threefry2x32(0u, 42u, c0, c1);
  return u01(threefry2x32(k.x, k.y, idx, 0u).x);
}

// ------------- TDM: stage the 128x5 f32 GT table into LDS (CDNA5 path) ------
__device__ __forceinline__ void tdm_load_gt_to_lds(const float* gt, void* ldsDst) {
  uint32_t lds = (uint32_t)(uintptr_t)ldsDst;   // low 32b of generic ptr = LDS offset
  uint64_t ga  = (uint64_t)(uintptr_t)gt;
  const uint32_t elems = NUM_GT * 5;            // 640 f32, 1-D tile
  u32x4 g0 = {1u,                               // count=1, user descriptor
              lds,                              // lds_addr
              (uint32_t)ga,                     // global_addr[31:0]
              (((uint32_t)(ga >> 32)) & 0x01FFFFFFu) | (2u << 30)}; // addr[56:32] | type=2
  i32x8 g1 = {(int)(2u << 16),                          // data_size=4B
              (int)((elems & 0xFFFFu) << 16),           // tensor_dim0[15:0]
              (int)(1u << 16),                          // tensor_dim0 hi=0 | tensor_dim1=1
              (int)((elems & 0xFFFFu) << 16),           // tile_dim0=640
              (int)1,                                   // tile_dim1=1, tile_dim2=0
              (int)elems,                               // tensor_dim0_stride lo
              (int)((elems & 0xFFFFu) << 16),           // stride0 hi=0 | stride1 lo
              0};                                       // stride1 hi
  i32x4 g2 = {0, 0, 0, 0};
  i32x4 g3 = {0, 0, 0, 0};
  i32x8 g4 = {0, 0, 0, 0, 0, 0, 0, 0};                  // extra group (clang-23 6-arg form)
  __builtin_amdgcn_tensor_load_to_lds(g0, g1, g2, g3, g4, 0);
  __builtin_amdgcn_s_wait_tensorcnt(0);
}

__device__ __forceinline__ float iou1(float ax1, float ay1, float ax2, float ay2, float aarea,
                                      float gx1, float gy1, float gx2, float gy2) {
  float iw = fminf(ax2, gx2) - fmaxf(ax1, gx1) + 1.0f; iw = fmaxf(iw, 0.0f);
  float ih = fminf(ay2, gy2) - fmaxf(ay1, gy1) + 1.0f; ih = fmaxf(ih, 0.0f);
  float inter = iw * ih;
  float garea = (gx2 - gx1 + 1.0f) * (gy2 - gy1 + 1.0f);
  return inter / (aarea + garea - inter);
}

// ---------------- K0: init workspace ----------------
__global__ void k_init(unsigned* gtMaxEnc, int* counters, int* hist) {
  int t = threadIdx.x;
  if (t < NUM_GT) gtMaxEnc[t] = fenc(-1.0f);
  if (t < 8) counters[t] = 0;
  for (int b = t; b < NBINS; b += blockDim.x) hist[b] = 0;
}

// ---------------- K1: IoU, per-anchor max/argmax, per-gt column max ----------
__global__ void k_iou(const float* __restrict__ anchors, const float* __restrict__ gt,
                      const float* __restrict__ iminfo,
                      float* __restrict__ maxov, int* __restrict__ argmax,
                      unsigned* __restrict__ gtMaxEnc) {
  __shared__ float sGT[NUM_GT * 5];
  __shared__ unsigned sMax[NUM_GT];
  if (threadIdx.x < 32) tdm_load_gt_to_lds(gt, sGT);     // TDM DMA, wave0 issues
  for (int g = threadIdx.x; g < NUM_GT; g += blockDim.x) sMax[g] = fenc(-1.0f);
  __syncthreads();

  int i = blockIdx.x * blockDim.x + threadIdx.x;
  float ax1 = anchors[4*i+0], ay1 = anchors[4*i+1];
  float ax2 = anchors[4*i+2], ay2 = anchors[4*i+3];
  bool inside = (ax1 >= 0.0f) & (ay1 >= 0.0f) & (ax2 < iminfo[1]) & (ay2 < iminfo[0]);
  float aarea = (ax2 - ax1 + 1.0f) * (ay2 - ay1 + 1.0f);

  float best = -1.0f; int bestg = 0; float bestMasked = -1.0f;
  for (int g = 0; g < NUM_GT; ++g) {
    float ov = iou1(ax1, ay1, ax2, ay2, aarea,
                    sGT[5*g+0], sGT[5*g+1], sGT[5*g+2], sGT[5*g+3]);
    if (ov > best) { best = ov; bestg = g; }            // argmax over RAW ov (reference)
    if (inside) {
      bestMasked = fmaxf(bestMasked, ov);
      atomicMax(&sMax[g], fenc(ov));                    // ds_max_u32 column max
    }
  }
  maxov[i]  = inside ? bestMasked : -1.0f;              // masked row max (mov)
  argmax[i] = bestg;
  __syncthreads();
  if (threadIdx.x < NUM_GT)
    atomicMax(&gtMaxEnc[threadIdx.x], sMax[threadIdx.x]); // global_atomic_max_u32
}

// ---------------- K2: labels + fg list + bg histogram ----------------
__global__ void k_labels(const float* __restrict__ anchors, const float* __restrict__ gt,
                         const float* __restrict__ iminfo,
                         const float* __restrict__ maxov, const unsigned* __restrict__ gtMaxEnc,
                         int* __restrict__ labelsOut, uint2* __restrict__ fgList,
                         int* __restrict__ counters, int* __restrict__ hist) {
  __shared__ float sGT[NUM_GT * 5];
  __shared__ float sMaxG[NUM_GT];
  if (threadIdx.x < 32) tdm_load_gt_to_lds(gt, sGT);
  for (int g = threadIdx.x; g < NUM_GT; g += blockDim.x) sMaxG[g] = fdec(gtMaxEnc[g]);
  __syncthreads();

  int i = blockIdx.x * blockDim.x + threadIdx.x;
  float ax1 = anchors[4*i+0], ay1 = anchors[4*i+1];
  float ax2 = anchors[4*i+2], ay2 = anchors[4*i+3];
  bool inside = (ax1 >= 0.0f) & (ay1 >= 0.0f) & (ax2 < iminfo[1]) & (ay2 < iminfo[0]);
  float aarea = (ax2 - ax1 + 1.0f) * (ay2 - ay1 + 1.0f);

  bool isbest = false;
  if (inside) {
    for (int g = 0; g < NUM_GT; ++g) {
      float ov = iou1(ax1, ay1, ax2, ay2, aarea,
                      sGT[5*g+0], sGT[5*g+1], sGT[5*g+2], sGT[5*g+3]);
      isbest |= (ov == sMaxG[g]);
    }
  }
  float mo = maxov[i];
  int label = -1;                                        // reference label logic, incl. quirk:
  if (inside && mo < RPN_NEG) label = 0;                 //  labels = -labels for is_best
  if (isbest) label = -label;                            //  (-0.0 == 0.0 downstream)
  if (inside && mo >= RPN_POS) label = 1;
  labelsOut[i] = label;

  if (label == 1) {
    float r = draw_uniform(0u, 1u, (uint32_t)i);         // fg subsample draw
    int pos = atomicAdd(&counters[CNT_FG], 1);
    if (pos < FG_CAP) fgList[pos] = make_uint2(fenc(r), (unsigned)i);
  } else if (label == 0) {
    float r = draw_uniform(2u, 3u, (uint32_t)i);         // bg subsample draw
    int bin = min((int)(r * (float)NBINS), NBINS - 1);
    atomicAdd(&hist[bin], 1);
  }
}

// ---------------- K3: exact fg selection (rank == argsort(argsort)) ---------
__global__ void k_fgsel(const uint2* __restrict__ fgList, int* __restrict__ counters,
                        int* __restrict__ labelsOut) {
  int F = min(counters[CNT_FG], FG_CAP);
  if (F > NUM_FG) {
    for (int t = threadIdx.x; t < F; t += blockDim.x) {
      uint2 me = fgList[t];
      int rank = 0;
      for (int j = 0; j < F; ++j) {                      // lists are L2-resident
        uint2 o = fgList[j];
        rank += (o.x < me.x) | ((o.x == me.x) & (o.y < me.y));
      }
      if (rank >= NUM_FG) labelsOut[me.y] = -1;
    }
  }
  if (threadIdx.x == 0) {
    int kept = min(F, NUM_FG);
    counters[CNT_FGKEPT] = kept;
    counters[CNT_NUMBG]  = RPN_BATCH - kept;
  }
}

// ---------------- K4: bg threshold bin from histogram ----------------
__global__ void k_bgthr(const int* __restrict__ hist, int* __restrict__ counters) {
  if (threadIdx.x == 0 && blockIdx.x == 0) {
    int numbg = counters[CNT_NUMBG];
    int cum = 0, b = NBINS, base = 0x7fffffff;
    for (int v = 0; v < NBINS; ++v) {
      int h = hist[v];
      if (cum + h >= numbg) { b = v; base = numbg - cum; break; }
      cum += h;
    }
    counters[CNT_BOUND] = b;                             // b==NBINS -> keep all bg
    counters[CNT_RANKBASE] = base;
  }
}

// ---------------- K5: mark bg by bin; collect boundary-bin ties -------------
__global__ void k_bgmark(int* __restrict__ labelsOut, const int* __restrict__ counters,
                         uint2* __restrict__ tieList, int* __restrict__ counters2) {
  int i = blockIdx.x * blockDim.x + threadIdx.x;
  if (labelsOut[i] != 0) return;
  int b = counters[CNT_BOUND];
  if (b >= NBINS) return;                                // all bg kept
  float r = draw_uniform(2u, 3u, (uint32_t)i);
  int bin = min((int)(r * (float)NBINS), NBINS - 1);
  if (bin > b) {
    labelsOut[i] = -1;
  } else if (bin == b) {
    int p = atomicAdd(&counters2[CNT_TIE], 1);
    if (p < TIE_CAP) tieList[p] = make_uint2(fenc(r), (unsigned)i);
  }
}

// ---------------- K6: exact rank within boundary bin ----------------
__global__ void k_tiesel(const uint2* __restrict__ tieList, const int* __restrict__ counters,
                         int* __restrict__ labelsOut) {
  int T = min(counters[CNT_TIE], TIE_CAP);
  int base = counters[CNT_RANKBASE];
  for (int t = threadIdx.x; t < T; t += blockDim.x) {
    uint2 me = tieList[t];
    int rank = 0;
    for (int j = 0; j < T; ++j) {
      uint2 o = tieList[j];
      rank += (o.x < me.x) | ((o.x == me.x) & (o.y < me.y));
    }
    if (rank >= base) labelsOut[me.y] = -1;
  }
}

// ---------------- K7: count examples (labels >= 0) ----------------
__global__ void k_count(const int* __restrict__ labelsOut, int* __restrict__ counters) {
  __shared__ int s;
  if (threadIdx.x == 0) s = 0;
  __syncthreads();
  int i = blockIdx.x * blockDim.x + threadIdx.x;
  atomicAdd(&s, (labelsOut[i] >= 0) ? 1 : 0);
  __syncthreads();
  if (threadIdx.x == 0) atomicAdd(&counters[CNT_NUMEX], s);
}

// ---------------- K8: emit targets + inside/outside weights ----------------
__global__ void k_out(const float* __restrict__ anchors, const float* __restrict__ gt,
                      const float* __restrict__ iminfo, const int* __restrict__ argmax,
                      const int* __restrict__ counters, const int* __restrict__ labelsOut,
                      float* __restrict__ out) {
  int i = blockIdx.x * blockDim.x + threadIdx.x;
  int label = labelsOut[i];
  float ax1 = anchors[4*i+0], ay1 = anchors[4*i+1];
  float ax2 = anchors[4*i+2], ay2 = anchors[4*i+3];
  bool inside = (ax1 >= 0.0f) & (ay1 >= 0.0f) & (ax2 < iminfo[1]) & (ay2 < iminfo[0]);

  int g = argmax[i];
  float gx1 = gt[5*g+0], gy1 = gt[5*g+1], gx2 = gt[5*g+2], gy2 = gt[5*g+3];
  float ew = ax2 - ax1 + 1.0f, eh = ay2 - ay1 + 1.0f;
  float ecx = ax1 + 0.5f * ew, ecy = ay1 + 0.5f * eh;
  float gw = gx2 - gx1 + 1.0f, gh = gy2 - gy1 + 1.0f;
  float gcx = gx1 + 0.5f * gw, gcy = gy1 + 0.5f * gh;
  float t0 = (gcx - ecx) / ew, t1 = (gcy - ecy) / eh;
  float t2 = logf(gw / ew),    t3 = logf(gh / eh);
  if (!inside) { t0 = t1 = t2 = t3 = 0.0f; }

  float* tgt = out + (size_t)N_ANCH;                     // rpn_bbox_targets
  tgt[4*i+0] = t0; tgt[4*i+1] = t1; tgt[4*i+2] = t2; tgt[4*i+3] = t3;

  float iw = (label == 1) ? 1.0f : 0.0f;
  float* insw = out + (size_t)5 * N_ANCH;                // inside weights
  insw[4*i+0] = iw; insw[4*i+1] = iw; insw[4*i+2] = iw; insw[4*i+3] = iw;

  float numex = (float)counters[CNT_NUMEX];
  float ow = ((label == 1) || (label == 0)) ? (1.0f / numex) : 0.0f;
  float* outw = out + (size_t)9 * N_ANCH;                // outside weights
  outw[4*i+0] = ow; outw[4*i+1] = ow; outw[4*i+2] = ow; outw[4*i+3] = ow;
}

extern "C" void kernel_launch(void* const* d_in, const int* in_sizes, int n_in,
                              void* d_out, int out_size, void* d_ws, size_t ws_size,
                              hipStream_t stream) {
  (void)in_sizes; (void)n_in; (void)out_size; (void)ws_size;
  // inputs (setup_inputs order): rpn_cls_score (unused except shape), gt_boxes, im_info, all_anchors
  const float* gt      = (const float*)d_in[1];
  const float* iminfo  = (const float*)d_in[2];
  const float* anchors = (const float*)d_in[3];

  char* ws = (char*)d_ws;
  unsigned* gtMaxEnc = (unsigned*)(ws + OFF_GTMAX);
  int*      counters = (int*)(ws + OFF_CNT);
  int*      hist     = (int*)(ws + OFF_HIST);
  float*    maxov    = (float*)(ws + OFF_MAXOV);
  int*      argmax   = (int*)(ws + OFF_ARG);
  uint2*    fgList   = (uint2*)(ws + OFF_FG);
  uint2*    tieList  = (uint2*)(ws + OFF_TIE);

  float* out    = (float*)d_out;
  int*   labels = (int*)d_out;                           // int32 labels region [0, N)

  dim3 blk(256), grid(NBLK);
  k_init  <<<1, 1024, 0, stream>>>(gtMaxEnc, counters, hist);
  k_iou   <<<grid, blk, 0, stream>>>(anchors, gt, iminfo, maxov, argmax, gtMaxEnc);
  k_labels<<<grid, blk, 0, stream>>>(anchors, gt, iminfo, maxov, gtMaxEnc,
                                     labels, fgList, counters, hist);
  k_fgsel <<<1, 256, 0, stream>>>(fgList, counters, labels);
  k_bgthr <<<1, 1, 0, stream>>>(hist, counters);
  k_bgmark<<<grid, blk, 0, stream>>>(labels, counters, tieList, counters);
  k_tiesel<<<1, 256, 0, stream>>>(tieList, counters, labels);
  k_count <<<grid, blk, 0, stream>>>(labels, counters);
  k_out   <<<grid, blk, 0, stream>>>(anchors, gt, iminfo, argmax, counters, labels, out);
}